// ProbAttention_51634096832752
// MI455X (gfx1250) — compile-verified
//
#include <hip/hip_runtime.h>
#include <math.h>

// Problem constants (from reference)
#define B_  2
#define L_  2048
#define HID_ 512
#define NH_ 8
#define HD_ 64
#define SK_ 40
#define NTOP_ 40
#define ROWS_ (B_ * L_)          // 4096

typedef __attribute__((ext_vector_type(2))) float v2f;
typedef __attribute__((ext_vector_type(8))) float v8f;
typedef __attribute__((ext_vector_type(4))) int  v4i;

#if __has_builtin(__builtin_amdgcn_global_load_async_to_lds_b128)
#define ASYNC_LDS 1
typedef __attribute__((address_space(1))) v4i gv4i;   // global 128-bit chunk
typedef __attribute__((address_space(3))) v4i lv4i;   // LDS 128-bit chunk
#endif

__device__ __forceinline__ void wait_asynccnt0() {
#if __has_builtin(__builtin_amdgcn_s_wait_asynccnt)
    __builtin_amdgcn_s_wait_asynccnt(0);
#else
    asm volatile("s_wait_asynccnt 0x0" ::: "memory");
#endif
}

// ---------------------------------------------------------------------------
// GEMM: out = A(4096x512) @ W(512x512)^T + bias, via V_WMMA_F32_16X16X4_F32.
// head_major=1 -> write out[b][h][l][d] (Q/K/V layout); else row-major [r][c].
// Workgroup: 256 threads = 8 waves; tile 64(M) x 128(N); K chunked by 64 in LDS.
// LDS staging uses GLOBAL_LOAD_ASYNC_TO_LDS_B128 (ASYNCcnt) when available.
// ---------------------------------------------------------------------------
#define TM 64
#define TN 128
#define TK 64
#define LDT 68   // padded LDS row stride (even -> 8B-aligned b64 frag loads)

__global__ __launch_bounds__(256)
void gemm_wmma_f32(const float* __restrict__ A, const float* __restrict__ W,
                   const float* __restrict__ bias, float* __restrict__ out,
                   int head_major)
{
    __shared__ float As[TM][LDT];
    __shared__ float Bs[TN][LDT];

    const int tid  = threadIdx.x;
    const int lane = tid & 31;          // wave32
    const int wave = tid >> 5;
    const int rowTile = blockIdx.y * TM;
    const int colTile = blockIdx.x * TN;
    const int m_sub = (wave & 3) * 16;  // wave's 16-row strip
    const int n_sub = (wave >> 2) * 64; // wave's 64-col strip (4 WMMA tiles)

    v8f acc[4] = {};

    const int laneN = lane & 15;
    const int kHi   = (lane >> 4) << 1; // 0 or 2 : K sub-offset per ISA layout
    const int aR    = m_sub + laneN;

    for (int kt = 0; kt < HID_; kt += TK) {
#ifdef ASYNC_LDS
        // Async DMA of both tiles into LDS (no VGPR round-trip, ASYNCcnt-tracked)
        #pragma unroll
        for (int i = 0; i < 4; ++i) {
            int chunk = tid + i * 256;
            int r  = chunk >> 4;
            int k4 = (chunk & 15) << 2;
            __builtin_amdgcn_global_load_async_to_lds_b128(
                (gv4i*)(A + (size_t)(rowTile + r) * HID_ + kt + k4),
                (lv4i*)&As[r][k4], 0, 0);
        }
        #pragma unroll
        for (int i = 0; i < 8; ++i) {
            int chunk = tid + i * 256;
            int c  = chunk >> 4;
            int k4 = (chunk & 15) << 2;
            __builtin_amdgcn_global_load_async_to_lds_b128(
                (gv4i*)(W + (size_t)(colTile + c) * HID_ + kt + k4),
                (lv4i*)&Bs[c][k4], 0, 0);
        }
        wait_asynccnt0();
#else
        // Fallback: VGPR-staged float4 copies
        #pragma unroll
        for (int i = 0; i < 4; ++i) {
            int chunk = tid + i * 256;
            int r  = chunk >> 4;
            int k4 = (chunk & 15) << 2;
            float4 t = *(const float4*)(A + (size_t)(rowTile + r) * HID_ + kt + k4);
            *(float4*)(&As[r][k4]) = t;
        }
        #pragma unroll
        for (int i = 0; i < 8; ++i) {
            int chunk = tid + i * 256;
            int c  = chunk >> 4;
            int k4 = (chunk & 15) << 2;
            float4 t = *(const float4*)(W + (size_t)(colTile + c) * HID_ + kt + k4);
            *(float4*)(&Bs[c][k4]) = t;
        }
#endif
        if (kt + TK < HID_) {
            // gfx1250 global_prefetch_b8 of next K chunk
            __builtin_prefetch(A + (size_t)(rowTile + (tid >> 2)) * HID_ + kt + TK, 0, 1);
            __builtin_prefetch(W + (size_t)(colTile + (tid >> 1)) * HID_ + kt + TK, 0, 1);
        }
        __syncthreads();

        #pragma unroll
        for (int kk = 0; kk < TK; kk += 4) {
            v2f a  = *(const v2f*)(&As[aR][kk + kHi]);
            v2f b0 = *(const v2f*)(&Bs[n_sub +  0 + laneN][kk + kHi]);
            v2f b1 = *(const v2f*)(&Bs[n_sub + 16 + laneN][kk + kHi]);
            v2f b2 = *(const v2f*)(&Bs[n_sub + 32 + laneN][kk + kHi]);
            v2f b3 = *(const v2f*)(&Bs[n_sub + 48 + laneN][kk + kHi]);
            acc[0] = __builtin_amdgcn_wmma_f32_16x16x4_f32(false, a, false, b0, (short)0, acc[0], false, false);
            acc[1] = __builtin_amdgcn_wmma_f32_16x16x4_f32(false, a, false, b1, (short)0, acc[1], false, false);
            acc[2] = __builtin_amdgcn_wmma_f32_16x16x4_f32(false, a, false, b2, (short)0, acc[2], false, false);
            acc[3] = __builtin_amdgcn_wmma_f32_16x16x4_f32(false, a, false, b3, (short)0, acc[3], false, false);
        }
        __syncthreads();
    }

    // Epilogue: D[j + 8*(lane>>4)][lane&15] per 16x16 f32 C/D layout
    const int laneM = (lane >> 4) << 3;
    #pragma unroll
    for (int t = 0; t < 4; ++t) {
        const int C = colTile + n_sub + t * 16 + laneN;
        const float bC = bias[C];
        #pragma unroll
        for (int j = 0; j < 8; ++j) {
            int R = rowTile + m_sub + j + laneM;
            float val = acc[t][j] + bC;
            if (head_major) {
                int b = R >> 11, l = R & (L_ - 1);
                int h = C >> 6,  d = C & (HD_ - 1);
                out[(((size_t)(b * NH_ + h)) * L_ + l) * HD_ + d] = val;
            } else {
                out[(size_t)R * HID_ + C] = val;
            }
        }
    }

#if __has_builtin(__builtin_amdgcn_s_cluster_barrier)
    // NOP when not dispatched in a cluster (ISA 3.2); exercises gfx1250 barrier path.
    __builtin_amdgcn_s_cluster_barrier();
#endif
}

// ---------------------------------------------------------------------------
// m[b][h][l] = max_s(q.k_samp) - sum_s(q.k_samp)/L    (one thread per row)
// ---------------------------------------------------------------------------
__global__ __launch_bounds__(256)
void sample_scores(const float* __restrict__ q, const float* __restrict__ k,
                   const int* __restrict__ idx, float* __restrict__ m)
{
    int gid = blockIdx.x * 256 + threadIdx.x;   // 0 .. 32767 = bh*L + l
    int l  = gid & (L_ - 1);
    int bh = gid >> 11;
    const float4* qr = (const float4*)(q + (size_t)gid * HD_);
    float4 qv[16];
    #pragma unroll
    for (int i = 0; i < 16; ++i) qv[i] = qr[i];
    float mx = -3.4e38f, sm = 0.f;
    for (int s = 0; s < SK_; ++s) {
        int ks = idx[l * SK_ + s];
        const float4* kr = (const float4*)(k + ((size_t)bh * L_ + ks) * HD_);
        float dot = 0.f;
        #pragma unroll
        for (int i = 0; i < 16; ++i) {
            float4 kv = kr[i];
            dot += qv[i].x * kv.x + qv[i].y * kv.y + qv[i].z * kv.z + qv[i].w * kv.w;
        }
        mx = fmaxf(mx, dot);
        sm += dot;
    }
    m[gid] = mx - sm * (1.0f / (float)L_);
}

// ---------------------------------------------------------------------------
// Top-40 indices of m[bh][:] by iterative argmax (lower index wins ties,
// matching jax.lax.top_k). One workgroup per (b,h).
// ---------------------------------------------------------------------------
__global__ __launch_bounds__(256)
void topk_kernel(const float* __restrict__ m, int* __restrict__ mtop)
{
    __shared__ float vals[L_];
    __shared__ float redV[256];
    __shared__ int   redI[256];
    int bh = blockIdx.x, tid = threadIdx.x;
    for (int i = tid; i < L_; i += 256) vals[i] = m[(size_t)bh * L_ + i];
    __syncthreads();
    for (int t = 0; t < NTOP_; ++t) {
        float bv = -3.4e38f; int bi = 0x7fffffff;
        for (int i = tid; i < L_; i += 256) {
            float v = vals[i];
            if (v > bv || (v == bv && i < bi)) { bv = v; bi = i; }
        }
        redV[tid] = bv; redI[tid] = bi;
        __syncthreads();
        for (int s = 128; s > 0; s >>= 1) {
            if (tid < s) {
                float ov = redV[tid + s]; int oi = redI[tid + s];
                if (ov > redV[tid] || (ov == redV[tid] && oi < redI[tid])) {
                    redV[tid] = ov; redI[tid] = oi;
                }
            }
            __syncthreads();
        }
        if (tid == 0) { mtop[bh * NTOP_ + t] = redI[0]; vals[redI[0]] = -3.4e38f; }
        __syncthreads();
    }
}

// ---------------------------------------------------------------------------
// vmean[bh][d] = mean_l v[bh][l][d]
// ---------------------------------------------------------------------------
__global__ __launch_bounds__(256)
void vmean_kernel(const float* __restrict__ v, float* __restrict__ vmean)
{
    __shared__ float part[4][HD_];
    int bh = blockIdx.x, tid = threadIdx.x;
    int d = tid & (HD_ - 1), g = tid >> 6;
    float s = 0.f;
    for (int l = g * 512; l < (g + 1) * 512; ++l)
        s += v[((size_t)bh * L_ + l) * HD_ + d];
    part[g][d] = s;
    __syncthreads();
    if (tid < HD_)
        vmean[bh * HD_ + d] = (part[0][d] + part[1][d] + part[2][d] + part[3][d]) * (1.0f / (float)L_);
}

// ---------------------------------------------------------------------------
// ctx[b][l][h*64+d] = vmean[b][h][d]  (broadcast fill, overwritten at top rows)
// ---------------------------------------------------------------------------
__global__ __launch_bounds__(256)
void bcast_ctx(const float* __restrict__ vmean, float* __restrict__ ctx)
{
    size_t gid = (size_t)blockIdx.x * 256 + threadIdx.x;  // B*L*HID elements
    int c = (int)(gid & (HID_ - 1));
    int b = (int)(gid >> 20);                             // L*HID = 2^20
    ctx[gid] = vmean[b * HID_ + c];
}

// ---------------------------------------------------------------------------
// Dense attention for the 40 selected queries per head. One workgroup per
// (b,h,i). Scores over all 2048 keys in LDS, block-reduced softmax, then
// d-parallel weighted-V accumulation; scatter into ctx.
// ---------------------------------------------------------------------------
__global__ __launch_bounds__(256)
void sel_attn(const float* __restrict__ q, const float* __restrict__ k,
              const float* __restrict__ v, const int* __restrict__ mtop,
              float* __restrict__ ctx)
{
    __shared__ float sc[L_];
    __shared__ float qi[HD_];
    __shared__ float red[256];
    int bh = blockIdx.x / NTOP_;
    int i  = blockIdx.x % NTOP_;
    int b = bh >> 3, h = bh & 7;
    int tid = threadIdx.x;
    int l0 = mtop[bh * NTOP_ + i];
    if (tid < HD_) qi[tid] = q[((size_t)bh * L_ + l0) * HD_ + tid] * 0.125f; // /sqrt(64)
    __syncthreads();

    float mx = -3.4e38f;
    for (int l = tid; l < L_; l += 256) {
        const float* kr = k + ((size_t)bh * L_ + l) * HD_;
        float dot = 0.f;
        #pragma unroll
        for (int d = 0; d < HD_; ++d) dot += qi[d] * kr[d];
        sc[l] = dot;
        mx = fmaxf(mx, dot);
    }
    red[tid] = mx; __syncthreads();
    for (int s = 128; s > 0; s >>= 1) { if (tid < s) red[tid] = fmaxf(red[tid], red[tid + s]); __syncthreads(); }
    float M = red[0];
    __syncthreads();

    float sm = 0.f;
    for (int l = tid; l < L_; l += 256) { float e = __expf(sc[l] - M); sc[l] = e; sm += e; }
    red[tid] = sm; __syncthreads();
    for (int s = 128; s > 0; s >>= 1) { if (tid < s) red[tid] += red[tid + s]; __syncthreads(); }
    float S = red[0];
    __syncthreads();

    if (tid < HD_) {
        float accv = 0.f;
        const float* vb = v + (size_t)bh * L_ * HD_ + tid;
        for (int l = 0; l < L_; ++l) accv += sc[l] * vb[(size_t)l * HD_];
        ctx[((size_t)b * L_ + l0) * HID_ + h * HD_ + tid] = accv / S;
    }
}

// ---------------------------------------------------------------------------
extern "C" void kernel_launch(void* const* d_in, const int* in_sizes, int n_in,
                              void* d_out, int out_size, void* d_ws, size_t ws_size,
                              hipStream_t stream)
{
    const float* hs  = (const float*)d_in[0];
    const int*   idx = (const int*)d_in[1];
    const float* Wq  = (const float*)d_in[2];
    const float* bq  = (const float*)d_in[3];
    const float* Wk  = (const float*)d_in[4];
    const float* bk  = (const float*)d_in[5];
    const float* Wv  = (const float*)d_in[6];
    const float* bv  = (const float*)d_in[7];
    const float* Wo  = (const float*)d_in[8];
    const float* bo  = (const float*)d_in[9];

    const size_t QKV = (size_t)B_ * NH_ * L_ * HD_;      // 2,097,152 floats
    float* ws    = (float*)d_ws;
    float* q     = ws;
    float* kbuf  = ws + QKV;
    float* vbuf  = ws + 2 * QKV;
    float* m     = ws + 3 * QKV;                         // B*NH*L
    float* vmean = m + (size_t)B_ * NH_ * L_;            // B*NH*HD
    float* ctx   = vmean + (size_t)B_ * NH_ * HD_;       // B*L*HID
    int*   mtop  = (int*)(ctx + (size_t)B_ * L_ * HID_); // B*NH*NTOP

    dim3 gGemm(HID_ / TN, ROWS_ / TM);   // (4, 64)
    dim3 bGemm(256);

    gemm_wmma_f32<<<gGemm, bGemm, 0, stream>>>(hs, Wq, bq, q,    1);
    gemm_wmma_f32<<<gGemm, bGemm, 0, stream>>>(hs, Wk, bk, kbuf, 1);
    gemm_wmma_f32<<<gGemm, bGemm, 0, stream>>>(hs, Wv, bv, vbuf, 1);

    sample_scores<<<(B_ * NH_ * L_) / 256, 256, 0, stream>>>(q, kbuf, idx, m);
    topk_kernel<<<B_ * NH_, 256, 0, stream>>>(m, mtop);
    vmean_kernel<<<B_ * NH_, 256, 0, stream>>>(vbuf, vmean);
    bcast_ctx<<<(B_ * L_ * HID_) / 256, 256, 0, stream>>>(vmean, ctx);
    sel_attn<<<B_ * NH_ * NTOP_, 256, 0, stream>>>(q, kbuf, vbuf, mtop, ctx);

    gemm_wmma_f32<<<gGemm, bGemm, 0, stream>>>(ctx, Wo, bo, (float*)d_out, 0);
}